// DependencyGNN_87574383165646
// MI455X (gfx1250) — compile-verified
//
#include <hip/hip_runtime.h>
#include <hip/hip_bf16.h>

// Problem constants (match reference)
#define BB   32
#define NN   2048
#define EE   2048
#define HH   4
#define CC   64
#define FF   256            // H*C = D_IN
#define MM   (BB * NN)      // 65536 rows in the batched GEMM
#define EN   (EE + NN)      // edges + self loops per graph

typedef float v2f __attribute__((ext_vector_type(2)));
typedef float v8f __attribute__((ext_vector_type(8)));

#define WMMA_F32(A, B, C) \
  __builtin_amdgcn_wmma_f32_16x16x4_f32(false, (A), false, (B), (short)0, (C), false, false)

// ---------------------------------------------------------------------------
// GEMM: Y[M,256] = X[M,256] @ W[256,256], fp32 exact via V_WMMA_F32_16X16X4_F32
// One wave -> 32(M) x 64(N) tile = 8 accumulators; K loop steps of 4.
// Per k-step: 2 A-fragment b64 loads + 8 B b32 loads feed 8 independent WMMAs.
// A 16x4 f32 layout: lane half (lane>>4) picks K pair, vgpr j picks K parity.
// B 4x16 f32 layout: vgpr j holds row K = k0 + 2*half + j striped over lanes.
// D 16x16 f32 layout: vgpr g holds row M = g + 8*half.
// ---------------------------------------------------------------------------
__global__ __launch_bounds__(256) void gat_gemm_k256_wmma(
    const float* __restrict__ X, const float* __restrict__ W,
    float* __restrict__ Y) {
  const int lane = threadIdx.x & 31;
  const int wave = threadIdx.x >> 5;
  const int tile = blockIdx.x * 8 + wave;   // (MM/32)*(256/64) tiles
  const int tm   = tile >> 2;               // 32-row M tile
  const int strip = tile & 3;               // 64-wide N strip
  const int half = lane >> 4;               // 0: lanes 0-15, 1: lanes 16-31
  const int r    = lane & 15;

  const float* xrow0 = X + (size_t)(tm * 32 + r) * FF;   // A rows, sub-tile 0
  const float* xrow1 = xrow0 + (size_t)16 * FF;          // A rows, sub-tile 1
  const float* wbase = W + strip * 64 + r;               // B column base

  v8f acc00 = {}, acc01 = {}, acc02 = {}, acc03 = {};
  v8f acc10 = {}, acc11 = {}, acc12 = {}, acc13 = {};
#pragma unroll 2
  for (int k0 = 0; k0 < FF; k0 += 4) {
    const int ka = k0 + half * 2;
    v2f a0; a0.x = xrow0[ka]; a0.y = xrow0[ka + 1];
    v2f a1; a1.x = xrow1[ka]; a1.y = xrow1[ka + 1];
    const float* wk = wbase + (size_t)ka * FF;
    v2f b0; b0.x = wk[0];  b0.y = wk[FF + 0];
    v2f b1; b1.x = wk[16]; b1.y = wk[FF + 16];
    v2f b2; b2.x = wk[32]; b2.y = wk[FF + 32];
    v2f b3; b3.x = wk[48]; b3.y = wk[FF + 48];
    acc00 = WMMA_F32(a0, b0, acc00);
    acc01 = WMMA_F32(a0, b1, acc01);
    acc02 = WMMA_F32(a0, b2, acc02);
    acc03 = WMMA_F32(a0, b3, acc03);
    acc10 = WMMA_F32(a1, b0, acc10);
    acc11 = WMMA_F32(a1, b1, acc11);
    acc12 = WMMA_F32(a1, b2, acc12);
    acc13 = WMMA_F32(a1, b3, acc13);
  }

  float* yb0 = Y + (size_t)(tm * 32 + 8 * half) * FF + strip * 64 + r;
  float* yb1 = yb0 + (size_t)16 * FF;
#pragma unroll
  for (int g = 0; g < 8; ++g) {
    yb0[(size_t)g * FF + 0]  = acc00[g];
    yb0[(size_t)g * FF + 16] = acc01[g];
    yb0[(size_t)g * FF + 32] = acc02[g];
    yb0[(size_t)g * FF + 48] = acc03[g];
    yb1[(size_t)g * FF + 0]  = acc10[g];
    yb1[(size_t)g * FF + 16] = acc11[g];
    yb1[(size_t)g * FF + 32] = acc12[g];
    yb1[(size_t)g * FF + 48] = acc13[g];
  }
}

// ---------------------------------------------------------------------------
// a_src/a_dst: one wave per (node, head); 64-wide dot via 2 loads + shfl tree.
// ---------------------------------------------------------------------------
__global__ __launch_bounds__(256) void gat_attn_coeff(
    const float* __restrict__ Hm, const float* __restrict__ att_s,
    const float* __restrict__ att_d, float* __restrict__ a_s,
    float* __restrict__ a_d, int total) {
  const int wid  = (int)((blockIdx.x * blockDim.x + threadIdx.x) >> 5);
  const int lane = threadIdx.x & 31;
  if (wid >= total) return;
  const int hh = wid & (HH - 1);
  const float* hp = Hm + (size_t)(wid >> 2) * FF + hh * CC;
  const float* as = att_s + hh * CC;
  const float* ad = att_d + hh * CC;
  float s1 = hp[lane] * as[lane] + hp[lane + 32] * as[lane + 32];
  float s2 = hp[lane] * ad[lane] + hp[lane + 32] * ad[lane + 32];
#pragma unroll
  for (int off = 16; off; off >>= 1) {
    s1 += __shfl_down(s1, off, 32);
    s2 += __shfl_down(s2, off, 32);
  }
  if (lane == 0) { a_s[wid] = s1; a_d[wid] = s2; }
}

// Monotone order-preserving float<->uint encoding for atomic max
__device__ __forceinline__ unsigned enc_f32(float v) {
  unsigned u = __float_as_uint(v);
  return (u & 0x80000000u) ? ~u : (u | 0x80000000u);
}
__device__ __forceinline__ float dec_f32(unsigned e) {
  return (e & 0x80000000u) ? __uint_as_float(e & 0x7FFFFFFFu)
                           : __uint_as_float(~e);
}

__device__ __forceinline__ void edge_sd(const int* src, const int* dst, int b,
                                        int e, int& s, int& d) {
  if (e < EE) { s = src[b * EE + e]; d = dst[b * EE + e]; }
  else        { s = d = e - EE; }
}

// Pass A: segment max of leaky_relu(a_src[s]+a_dst[d]) into encoded-uint buffer
__global__ __launch_bounds__(256) void gat_edge_max(
    const int* __restrict__ src, const int* __restrict__ dst,
    const float* __restrict__ a_s, const float* __restrict__ a_d,
    unsigned* __restrict__ menc) {
  const int idx = blockIdx.x * blockDim.x + threadIdx.x;
  if (idx >= BB * EN) return;
  const int b = idx / EN, e = idx % EN;
  int s, d; edge_sd(src, dst, b, e, s, d);
  const float* as = a_s + (size_t)(b * NN + s) * HH;
  const float* ad = a_d + (size_t)(b * NN + d) * HH;
  unsigned* mb = menc + (size_t)(b * NN + d) * HH;
#pragma unroll
  for (int h = 0; h < HH; ++h) {
    float v = as[h] + ad[h];
    v = v > 0.0f ? v : 0.2f * v;       // leaky_relu slope 0.2
    atomicMax(mb + h, enc_f32(v));
  }
}

// Pass B: ex = exp(e - max); store per-edge, atomicAdd into denom
__global__ __launch_bounds__(256) void gat_edge_expsum(
    const int* __restrict__ src, const int* __restrict__ dst,
    const float* __restrict__ a_s, const float* __restrict__ a_d,
    const unsigned* __restrict__ menc, float* __restrict__ denom,
    float* __restrict__ ex) {
  const int idx = blockIdx.x * blockDim.x + threadIdx.x;
  if (idx >= BB * EN) return;
  const int b = idx / EN, e = idx % EN;
  int s, d; edge_sd(src, dst, b, e, s, d);
  const float* as = a_s + (size_t)(b * NN + s) * HH;
  const float* ad = a_d + (size_t)(b * NN + d) * HH;
  const unsigned* mb = menc + (size_t)(b * NN + d) * HH;
  float* db = denom + (size_t)(b * NN + d) * HH;
#pragma unroll
  for (int h = 0; h < HH; ++h) {
    float v = as[h] + ad[h];
    v = v > 0.0f ? v : 0.2f * v;
    float exv = __expf(v - dec_f32(mb[h]));
    ex[(size_t)idx * HH + h] = exv;
    atomicAdd(db + h, exv);
  }
}

// Pass C: out[d] += alpha * h[s]; one wave per edge, lanes cover C=64 per head.
__global__ __launch_bounds__(256) void gat_edge_agg(
    const int* __restrict__ src, const int* __restrict__ dst,
    const float* __restrict__ ex, const float* __restrict__ denom,
    const float* __restrict__ Hm, float* __restrict__ out) {
  const int wid  = (int)((blockIdx.x * blockDim.x + threadIdx.x) >> 5);
  const int lane = threadIdx.x & 31;
  if (wid >= BB * EN) return;
  const int b = wid / EN, e = wid % EN;
  int s, d; edge_sd(src, dst, b, e, s, d);
  const float* hrow = Hm + (size_t)(b * NN + s) * FF;
  float* orow = out + (size_t)(b * NN + d) * FF;
#pragma unroll
  for (int h = 0; h < HH; ++h) {
    float alpha = ex[(size_t)wid * HH + h] /
                  (denom[(size_t)(b * NN + d) * HH + h] + 1e-16f);
    atomicAdd(orow + h * CC + lane,      alpha * hrow[h * CC + lane]);
    atomicAdd(orow + h * CC + lane + 32, alpha * hrow[h * CC + lane + 32]);
  }
}

// Initialize aggregation buffer with the (broadcast) bias vector
__global__ __launch_bounds__(256) void gat_init_bias(
    float* __restrict__ out, const float* __restrict__ bias, int total) {
  const int i = blockIdx.x * blockDim.x + threadIdx.x;
  if (i < total) out[i] = bias[i & (FF - 1)];
}

// ELU (alpha = 1)
__global__ __launch_bounds__(256) void gat_elu(
    const float* __restrict__ in, float* __restrict__ out, int total) {
  const int i = blockIdx.x * blockDim.x + threadIdx.x;
  if (i < total) {
    float v = in[i];
    out[i] = v > 0.0f ? v : expm1f(v);
  }
}

extern "C" void kernel_launch(void* const* d_in, const int* in_sizes, int n_in,
                              void* d_out, int out_size, void* d_ws, size_t ws_size,
                              hipStream_t stream) {
  const float* x      = (const float*)d_in[0];   // [B,N,256]
  const int*   src    = (const int*)  d_in[1];   // [B,E]
  const int*   dst    = (const int*)  d_in[2];   // [B,E]
  const float* W1     = (const float*)d_in[3];   // [256,256]
  const float* att_s1 = (const float*)d_in[4];   // [H,C]
  const float* att_d1 = (const float*)d_in[5];
  const float* bias1  = (const float*)d_in[6];   // [256]
  const float* W2     = (const float*)d_in[7];
  const float* att_s2 = (const float*)d_in[8];
  const float* att_d2 = (const float*)d_in[9];
  const float* bias2  = (const float*)d_in[10];
  float* out = (float*)d_out;                    // [B,N,256]

  // Workspace layout (~140 MB)
  char* ws = (char*)d_ws;
  size_t off = 0;
  float*    bufA  = (float*)(ws + off); off += (size_t)MM * FF * 4;  // h1 / x2
  float*    bufB  = (float*)(ws + off); off += (size_t)MM * FF * 4;  // agg1 / h2
  float*    a_s   = (float*)(ws + off); off += (size_t)MM * HH * 4;
  float*    a_d   = (float*)(ws + off); off += (size_t)MM * HH * 4;
  unsigned* menc  = (unsigned*)(ws + off); off += (size_t)MM * HH * 4;
  float*    denom = (float*)(ws + off); off += (size_t)MM * HH * 4;
  float*    ex    = (float*)(ws + off); off += (size_t)BB * EN * HH * 4;

  const int gemmBlocks = (MM / 32) * (FF / 64) / 8;          // 1024
  const int attnBlocks = (MM * HH) * 32 / 256;               // 32768
  const int edgeBlocks = (BB * EN + 255) / 256;              // 512
  const int aggBlocks  = (BB * EN) * 32 / 256;               // 16384
  const int elemBlocks = (MM * FF) / 256;                    // 65536

  // -------- Layer 1 --------
  gat_gemm_k256_wmma<<<gemmBlocks, 256, 0, stream>>>(x, W1, bufA);
  gat_attn_coeff<<<attnBlocks, 256, 0, stream>>>(bufA, att_s1, att_d1, a_s, a_d, MM * HH);
  hipMemsetAsync(menc, 0, (size_t)MM * HH * 4, stream);   // enc(0)=below -inf
  hipMemsetAsync(denom, 0, (size_t)MM * HH * 4, stream);
  gat_init_bias<<<elemBlocks, 256, 0, stream>>>(bufB, bias1, MM * FF);
  gat_edge_max<<<edgeBlocks, 256, 0, stream>>>(src, dst, a_s, a_d, menc);
  gat_edge_expsum<<<edgeBlocks, 256, 0, stream>>>(src, dst, a_s, a_d, menc, denom, ex);
  gat_edge_agg<<<aggBlocks, 256, 0, stream>>>(src, dst, ex, denom, bufA, bufB);
  gat_elu<<<elemBlocks, 256, 0, stream>>>(bufB, bufA, MM * FF);  // x2 = elu(agg1)

  // -------- Layer 2 --------
  gat_gemm_k256_wmma<<<gemmBlocks, 256, 0, stream>>>(bufA, W2, bufB);  // h2
  gat_attn_coeff<<<attnBlocks, 256, 0, stream>>>(bufB, att_s2, att_d2, a_s, a_d, MM * HH);
  hipMemsetAsync(menc, 0, (size_t)MM * HH * 4, stream);
  hipMemsetAsync(denom, 0, (size_t)MM * HH * 4, stream);
  gat_init_bias<<<elemBlocks, 256, 0, stream>>>(out, bias2, MM * FF);
  gat_edge_max<<<edgeBlocks, 256, 0, stream>>>(src, dst, a_s, a_d, menc);
  gat_edge_expsum<<<edgeBlocks, 256, 0, stream>>>(src, dst, a_s, a_d, menc, denom, ex);
  gat_edge_agg<<<aggBlocks, 256, 0, stream>>>(src, dst, ex, denom, bufB, out);
}